// MolGATEncoder_12867722018910
// MI455X (gfx1250) — compile-verified
//
#include <hip/hip_runtime.h>
#include <math.h>
#include <float.h>

#define HIDD 256
#define NHEADS 4
#define HD 1024            // HEADS*HID
#define NNODE 50000
#define NEDGE 100000
#define NEF (NEDGE + NNODE)
#define NGR 1024
#define OUTD 768

typedef float v2f __attribute__((ext_vector_type(2)));
typedef float v8f __attribute__((ext_vector_type(8)));

struct P9 { const float* p[9]; };
struct P3 { const float* p[3]; };

// ---------------- float atomic max via int trick (NaN-free data) -------------
__device__ inline void atomicMaxF(float* addr, float val) {
    if (val >= 0.0f) atomicMax((int*)addr, __float_as_int(val));
    else             atomicMin((unsigned int*)addr, __float_as_uint(val));
}

__device__ inline float blockReduceSum(float v, float* red) {
    int tid = threadIdx.x;
    red[tid] = v; __syncthreads();
    for (int s = 128; s > 0; s >>= 1) {
        if (tid < s) red[tid] += red[tid + s];
        __syncthreads();
    }
    float r = red[0]; __syncthreads();
    return r;
}

// ============================================================================
// f32 WMMA GEMM, LDS double-buffered: C[M,NN] = act(A[M,KK] @ B[KK,NN] + bias)
// grid = (NN/128, ceil(M/64)), block = 256 (8 waves).
// Block tile 64(M) x 128(N); wave tile 16x64 (4 f32 WMMA accumulators).
// A staged [64][16] row-major (stride 18); B staged transposed [128][16]
// (stride 18) so each lane's (k,k+1) pair is a single ds_load_b64.
// ============================================================================
template <int KK, int NN>
__global__ __launch_bounds__(256) void k_gemm_f32(
    const float* __restrict__ A, const float* __restrict__ B,
    const float* __restrict__ bias, float* __restrict__ C,
    int M, int act)
{
    constexpr int NB = KK / 16;           // K-slabs
    __shared__ float lsA[2][64 * 18];     // 2 x 4608 B
    __shared__ float lsB[2][128 * 18];    // 2 x 9216 B

    int tid  = threadIdx.x;
    int wave = tid >> 5;
    int lane = tid & 31;
    int half = lane >> 4;
    int r    = lane & 15;
    int mtile = (wave & 3) << 4;          // 0,16,32,48
    int ntile = (wave >> 2) << 6;         // 0,64
    int mblk  = blockIdx.y << 6;
    int nblk  = blockIdx.x << 7;

    // global->LDS staging indices
    int arow = tid >> 2;                  // 0..63
    int akk  = (tid & 3) << 2;            // 0,4,8,12
    int agrow = mblk + arow;
    if (agrow >= M) agrow = M - 1;        // clamp (stores are guarded)
    const float* Ag = A + (size_t)agrow * KK + akk;

    int bk  = tid >> 4;                   // 0..15
    int bn8 = (tid & 15) << 3;            // 0..120
    const float* Bg = B + (size_t)bk * NN + nblk + bn8;

    float4 ra, rb0, rb1;
    ra  = *(const float4*)(Ag);
    rb0 = *(const float4*)(Bg);
    rb1 = *(const float4*)(Bg + 4);

    v8f acc[4] = {};

    for (int kb = 0; kb < NB; ++kb) {
        int buf = kb & 1;
        // stage current slab into LDS
        {
            float* pa = &lsA[buf][arow * 18 + akk];   // 8B aligned
            ((float2*)pa)[0] = make_float2(ra.x, ra.y);
            ((float2*)pa)[1] = make_float2(ra.z, ra.w);
            float* pb = &lsB[buf][0];
            pb[(bn8 + 0) * 18 + bk] = rb0.x;
            pb[(bn8 + 1) * 18 + bk] = rb0.y;
            pb[(bn8 + 2) * 18 + bk] = rb0.z;
            pb[(bn8 + 3) * 18 + bk] = rb0.w;
            pb[(bn8 + 4) * 18 + bk] = rb1.x;
            pb[(bn8 + 5) * 18 + bk] = rb1.y;
            pb[(bn8 + 6) * 18 + bk] = rb1.z;
            pb[(bn8 + 7) * 18 + bk] = rb1.w;
        }
        __syncthreads();
        // prefetch next slab (global latency overlaps the 16 WMMAs below)
        if (kb + 1 < NB) {
            int k0 = (kb + 1) << 4;
            ra  = *(const float4*)(Ag + k0);
            rb0 = *(const float4*)(Bg + (size_t)k0 * NN);
            rb1 = *(const float4*)(Bg + (size_t)k0 * NN + 4);
        }
        // compute on staged slab
#pragma unroll
        for (int kq = 0; kq < 4; ++kq) {
            int k = (kq << 2) + 2 * half;                       // even -> 8B aligned
            float2 af = *(const float2*)&lsA[buf][(mtile + r) * 18 + k];
            v2f a; a.x = af.x; a.y = af.y;
#pragma unroll
            for (int t = 0; t < 4; ++t) {
                float2 bf = *(const float2*)&lsB[buf][(ntile + (t << 4) + r) * 18 + k];
                v2f b; b.x = bf.x; b.y = bf.y;
                acc[t] = __builtin_amdgcn_wmma_f32_16x16x4_f32(
                    false, a, false, b, (short)0, acc[t], false, false);
            }
        }
        __syncthreads();
    }

    // epilogue: bias / activation / guarded store
#pragma unroll
    for (int t = 0; t < 4; ++t) {
        int col = nblk + ntile + (t << 4) + r;
        float bv = bias ? bias[col] : 0.0f;
#pragma unroll
        for (int v = 0; v < 8; ++v) {
            int grow = mblk + mtile + v + 8 * half;
            if (grow < M) {
                float val = acc[t][v] + bv;
                if (act == 1) val = fmaxf(val, 0.0f);
                C[(size_t)grow * NN + col] = val;
            }
        }
    }
}

// ---------------- embeddings ------------------------------------------------
__global__ void k_embed_nodes(const int* __restrict__ x, P9 tabs, float* __restrict__ h) {
    size_t idx = (size_t)blockIdx.x * blockDim.x + threadIdx.x;
    if (idx >= (size_t)NNODE * HIDD) return;
    int n = (int)(idx >> 8), c = (int)(idx & 255);
    const int* xr = x + (size_t)n * 9;
    float s = 0.0f;
#pragma unroll
    for (int i = 0; i < 9; ++i) s += tabs.p[i][(size_t)xr[i] * HIDD + c];
    h[idx] = s;
}

__global__ void k_embed_edges(const int* __restrict__ ea, P3 tabs, float* __restrict__ ef) {
    size_t idx = (size_t)blockIdx.x * blockDim.x + threadIdx.x;
    if (idx >= (size_t)NEDGE * HIDD) return;
    int j = (int)(idx >> 8), c = (int)(idx & 255);
    const int* er = ea + (size_t)j * 3;
    float s = 0.0f;
#pragma unroll
    for (int i = 0; i < 3; ++i) s += tabs.p[i][(size_t)er[i] * HIDD + c];
    ef[idx] = s;
}

// ---------------- self-loop edge features: scatter-mean of e over dst -------
__global__ void k_loop_accum(const int* __restrict__ ei, float* __restrict__ ef,
                             float* __restrict__ cnt) {
    size_t idx = (size_t)blockIdx.x * blockDim.x + threadIdx.x;
    if (idx >= (size_t)NEDGE * HIDD) return;
    int j = (int)(idx >> 8), c = (int)(idx & 255);
    int d = ei[NEDGE + j];
    atomicAdd(&ef[(size_t)(NEDGE + d) * HIDD + c], ef[(size_t)j * HIDD + c]);
    if (c == 0) atomicAdd(&cnt[d], 1.0f);
}

__global__ void k_loop_div(const float* __restrict__ cnt, float* __restrict__ ef) {
    size_t idx = (size_t)blockIdx.x * blockDim.x + threadIdx.x;
    if (idx >= (size_t)NNODE * HIDD) return;
    int n = (int)(idx >> 8), c = (int)(idx & 255);
    ef[(size_t)(NEDGE + n) * HIDD + c] /= fmaxf(cnt[n], 1.0f);
}

// ---------------- Wae[k,h] = sum_c We[k, h*256+c] * ae[h,c] -----------------
__global__ void k_wae(const float* __restrict__ We, const float* __restrict__ ae,
                      float* __restrict__ wae) {
    int t = blockIdx.x * blockDim.x + threadIdx.x;
    if (t >= HIDD * NHEADS) return;
    int k = t >> 2, hh = t & 3;
    float s = 0.0f;
    for (int c = 0; c < HIDD; ++c)
        s += We[(size_t)k * HD + hh * HIDD + c] * ae[hh * HIDD + c];
    wae[k * 4 + hh] = s;
}

// ---------------- a_e[j,h] = ef[j,:] . Wae[:,h]  (one wave per row) ---------
__global__ __launch_bounds__(256) void k_a_e(const float* __restrict__ ef,
                                             const float* __restrict__ wae,
                                             float* __restrict__ a_e) {
    int wave = threadIdx.x >> 5, lane = threadIdx.x & 31;
    size_t j = (size_t)blockIdx.x * 8 + wave;
    if (j >= NEF) return;
    const float* row = ef + j * HIDD;
    float acc[4] = {0.f, 0.f, 0.f, 0.f};
    for (int i = 0; i < 8; ++i) {
        int k = lane + 32 * i;
        float v = row[k];
#pragma unroll
        for (int hh = 0; hh < 4; ++hh) acc[hh] += v * wae[k * 4 + hh];
    }
#pragma unroll
    for (int hh = 0; hh < 4; ++hh) {
        float s = acc[hh];
        for (int off = 16; off > 0; off >>= 1) s += __shfl_down(s, off, 32);
        if (lane == 0) a_e[j * 4 + hh] = s;
    }
}

// ---------------- a_src/a_dst per node from xw (one wave per node) ----------
__global__ __launch_bounds__(256) void k_att_node(const float* __restrict__ xw,
                                                  const float* __restrict__ as_,
                                                  const float* __restrict__ ad_,
                                                  float* __restrict__ a_src,
                                                  float* __restrict__ a_dst) {
    int wave = threadIdx.x >> 5, lane = threadIdx.x & 31;
    size_t n = (size_t)blockIdx.x * 8 + wave;
    if (n >= NNODE) return;
#pragma unroll
    for (int hh = 0; hh < 4; ++hh) {
        float ss = 0.f, sd = 0.f;
        for (int i = 0; i < 8; ++i) {
            int c = lane + 32 * i;
            float v = xw[n * HD + hh * HIDD + c];
            ss += v * as_[hh * HIDD + c];
            sd += v * ad_[hh * HIDD + c];
        }
        for (int off = 16; off > 0; off >>= 1) {
            ss += __shfl_down(ss, off, 32);
            sd += __shfl_down(sd, off, 32);
        }
        if (lane == 0) { a_src[n * 4 + hh] = ss; a_dst[n * 4 + hh] = sd; }
    }
}

__global__ void k_fill(float* __restrict__ p, float v, size_t n) {
    size_t i = (size_t)blockIdx.x * blockDim.x + threadIdx.x;
    if (i < n) p[i] = v;
}

// ---------------- attention logits + segment max ----------------------------
__global__ void k_alpha_max(const int* __restrict__ ei,
                            const float* __restrict__ a_src, const float* __restrict__ a_dst,
                            const float* __restrict__ a_e,
                            float* __restrict__ alpha, float* __restrict__ mmax) {
    size_t j = (size_t)blockIdx.x * blockDim.x + threadIdx.x;
    if (j >= NEF) return;
    int s, d;
    if (j < NEDGE) { s = ei[j]; d = ei[NEDGE + j]; } else { s = d = (int)(j - NEDGE); }
#pragma unroll
    for (int hh = 0; hh < 4; ++hh) {
        float raw = a_src[(size_t)s * 4 + hh] + a_dst[(size_t)d * 4 + hh] + a_e[j * 4 + hh];
        raw = raw > 0.0f ? raw : 0.2f * raw;   // leaky_relu(0.2)
        alpha[j * 4 + hh] = raw;
        atomicMaxF(&mmax[(size_t)d * 4 + hh], raw);
    }
}

// ---------------- exp and segment sum ---------------------------------------
__global__ void k_alpha_exp(const int* __restrict__ ei, float* __restrict__ alpha,
                            const float* __restrict__ mmax, float* __restrict__ denom) {
    size_t j = (size_t)blockIdx.x * blockDim.x + threadIdx.x;
    if (j >= NEF) return;
    int d = (j < NEDGE) ? ei[NEDGE + j] : (int)(j - NEDGE);
#pragma unroll
    for (int hh = 0; hh < 4; ++hh) {
        float a = expf(alpha[j * 4 + hh] - mmax[(size_t)d * 4 + hh]);
        alpha[j * 4 + hh] = a;
        atomicAdd(&denom[(size_t)d * 4 + hh], a);
    }
}

// ---------------- weighted message scatter (one block per edge) -------------
__global__ __launch_bounds__(256) void k_message(const int* __restrict__ ei,
                                                 const float* __restrict__ alpha,
                                                 const float* __restrict__ denom,
                                                 const float* __restrict__ xw,
                                                 float* __restrict__ hout) {
    __shared__ int sd[2];
    __shared__ float w[4];
    size_t j = blockIdx.x;
    int tid = threadIdx.x;
    if (tid < 4) {
        int s, d;
        if (j < NEDGE) { s = ei[j]; d = ei[NEDGE + j]; } else { s = d = (int)(j - NEDGE); }
        if (tid == 0) { sd[0] = s; sd[1] = d; }
        w[tid] = alpha[j * 4 + tid] / (denom[(size_t)d * 4 + tid] + 1e-16f);
    }
    __syncthreads();
    const float* xs = xw + (size_t)sd[0] * HD;
    float* hd = hout + (size_t)sd[1] * HD;
#pragma unroll
    for (int q = 0; q < 4; ++q) {
        int c = q * 256 + tid;
        atomicAdd(&hd[c], xs[c] * w[q]);
    }
}

// ---------------- bias + LayerNorm + ELU (one block per node) ---------------
__global__ __launch_bounds__(256) void k_ln_elu(float* __restrict__ h,
                                                const float* __restrict__ b,
                                                const float* __restrict__ lng,
                                                const float* __restrict__ lnb) {
    __shared__ float red[256];
    size_t n = blockIdx.x;
    int tid = threadIdx.x;
    float xv[4];
    float s = 0.0f;
#pragma unroll
    for (int q = 0; q < 4; ++q) {
        int c = q * 256 + tid;
        float v = h[n * HD + c] + b[c];
        xv[q] = v; s += v;
    }
    float tot = blockReduceSum(s, red);
    float mu = tot * (1.0f / HD);
    float s2 = 0.0f;
#pragma unroll
    for (int q = 0; q < 4; ++q) { float dlt = xv[q] - mu; s2 += dlt * dlt; }
    float tot2 = blockReduceSum(s2, red);
    float rstd = rsqrtf(tot2 * (1.0f / HD) + 1e-5f);
#pragma unroll
    for (int q = 0; q < 4; ++q) {
        int c = q * 256 + tid;
        float y = (xv[q] - mu) * rstd * lng[c] + lnb[c];
        h[n * HD + c] = y > 0.0f ? y : (expf(y) - 1.0f);   // ELU
    }
}

// ---------------- graph pooling ---------------------------------------------
__global__ void k_pool(const float* __restrict__ hn, const int* __restrict__ batch,
                       float* __restrict__ gsum, float* __restrict__ gmax,
                       float* __restrict__ gcnt) {
    size_t idx = (size_t)blockIdx.x * blockDim.x + threadIdx.x;
    if (idx >= (size_t)NNODE * HIDD) return;
    int n = (int)(idx >> 8), c = (int)(idx & 255);
    int b = batch[n];
    float v = hn[idx];
    atomicAdd(&gsum[(size_t)b * HIDD + c], v);
    atomicMaxF(&gmax[(size_t)b * HIDD + c], v);
    if (c == 0) atomicAdd(&gcnt[b], 1.0f);
}

__global__ void k_pool_fin(const float* __restrict__ gsum, const float* __restrict__ gmax,
                           const float* __restrict__ gcnt, float* __restrict__ gcat) {
    size_t idx = (size_t)blockIdx.x * blockDim.x + threadIdx.x;
    if (idx >= (size_t)NGR * HIDD) return;
    int g = (int)(idx >> 8), c = (int)(idx & 255);
    gcat[(size_t)g * (2 * HIDD) + c] = gsum[idx] / fmaxf(gcnt[g], 1.0f);
    float mx = gmax[idx];
    gcat[(size_t)g * (2 * HIDD) + HIDD + c] = (mx > -FLT_MAX) ? mx : 0.0f;
}

// ---------------- L2 normalize rows of g (one block per graph) --------------
__global__ __launch_bounds__(256) void k_l2norm(float* __restrict__ g) {
    __shared__ float red[256];
    size_t r = blockIdx.x;
    int tid = threadIdx.x;
    float vv[3];
    float s = 0.0f;
#pragma unroll
    for (int q = 0; q < 3; ++q) {
        vv[q] = g[r * OUTD + q * 256 + tid];
        s += vv[q] * vv[q];
    }
    float tot = blockReduceSum(s, red);
    float sc = 1.0f / fmaxf(sqrtf(tot), 1e-12f);
#pragma unroll
    for (int q = 0; q < 3; ++q) g[r * OUTD + q * 256 + tid] = vv[q] * sc;
}

// ============================================================================
extern "C" void kernel_launch(void* const* d_in, const int* in_sizes, int n_in,
                              void* d_out, int out_size, void* d_ws, size_t ws_size,
                              hipStream_t stream) {
    const int* x     = (const int*)d_in[0];
    const int* ei    = (const int*)d_in[1];
    const int* ea    = (const int*)d_in[2];
    const int* batch = (const int*)d_in[3];
    P9 nt; for (int i = 0; i < 9; ++i) nt.p[i] = (const float*)d_in[4 + i];
    P3 et; for (int i = 0; i < 3; ++i) et.p[i] = (const float*)d_in[13 + i];

    float* g_out  = (float*)d_out;                          // [NGR, OUTD]
    float* h_node = (float*)d_out + (size_t)NGR * OUTD;     // [N, HIDD]

    // workspace carve
    float* w = (float*)d_ws;
    float* h     = w; w += (size_t)NNODE * HD;
    float* xw    = w; w += (size_t)NNODE * HD;
    float* ef    = w; w += (size_t)NEF * HIDD;
    float* a_src = w; w += (size_t)NNODE * 4;
    float* a_dst = w; w += (size_t)NNODE * 4;
    float* a_e   = w; w += (size_t)NEF * 4;
    float* alpha = w; w += (size_t)NEF * 4;
    float* mmax  = w; w += (size_t)NNODE * 4;
    float* denom = w; w += (size_t)NNODE * 4;
    float* cnt   = w; w += (size_t)NNODE;
    float* wae   = w; w += (size_t)HIDD * 4;
    float* gsum  = w; w += (size_t)NGR * HIDD;
    float* gmax  = w; w += (size_t)NGR * HIDD;
    float* gcnt  = w; w += (size_t)NGR;
    float* gcat  = w; w += (size_t)NGR * 2 * HIDD;
    float* g1    = w; w += (size_t)NGR * HIDD;

    // --- embeddings + self-loop edge features ---
    {
        size_t tot = (size_t)NNODE * HIDD;
        k_embed_nodes<<<(tot + 255) / 256, 256, 0, stream>>>(x, nt, h);
    }
    {
        size_t tot = (size_t)NEDGE * HIDD;
        k_embed_edges<<<(tot + 255) / 256, 256, 0, stream>>>(ea, et, ef);
    }
    hipMemsetAsync(ef + (size_t)NEDGE * HIDD, 0, (size_t)NNODE * HIDD * sizeof(float), stream);
    hipMemsetAsync(cnt, 0, (size_t)NNODE * sizeof(float), stream);
    k_loop_accum<<<((size_t)NEDGE * HIDD + 255) / 256, 256, 0, stream>>>(ei, ef, cnt);
    k_loop_div<<<((size_t)NNODE * HIDD + 255) / 256, 256, 0, stream>>>(cnt, ef);

    // --- GAT layers ---
    for (int l = 0; l < 3; ++l) {
        const float* W   = (const float*)d_in[16 + l * 8 + 0];
        const float* We  = (const float*)d_in[16 + l * 8 + 1];
        const float* as_ = (const float*)d_in[16 + l * 8 + 2];
        const float* ad_ = (const float*)d_in[16 + l * 8 + 3];
        const float* ae_ = (const float*)d_in[16 + l * 8 + 4];
        const float* bb  = (const float*)d_in[16 + l * 8 + 5];
        const float* lng = (const float*)d_in[16 + l * 8 + 6];
        const float* lnb = (const float*)d_in[16 + l * 8 + 7];

        // xw = h @ W  (f32 WMMA, LDS double-buffered)
        {
            dim3 grid(HD / 128, (NNODE + 63) / 64);
            if (l == 0)
                k_gemm_f32<HIDD, HD><<<grid, 256, 0, stream>>>(h, W, nullptr, xw, NNODE, 0);
            else
                k_gemm_f32<HD, HD><<<grid, 256, 0, stream>>>(h, W, nullptr, xw, NNODE, 0);
        }
        k_att_node<<<(NNODE + 7) / 8, 256, 0, stream>>>(xw, as_, ad_, a_src, a_dst);
        k_wae<<<(HIDD * 4 + 255) / 256, 256, 0, stream>>>(We, ae_, wae);
        k_a_e<<<(NEF + 7) / 8, 256, 0, stream>>>(ef, wae, a_e);

        k_fill<<<((size_t)NNODE * 4 + 255) / 256, 256, 0, stream>>>(mmax, -FLT_MAX, (size_t)NNODE * 4);
        hipMemsetAsync(denom, 0, (size_t)NNODE * 4 * sizeof(float), stream);
        k_alpha_max<<<(NEF + 255) / 256, 256, 0, stream>>>(ei, a_src, a_dst, a_e, alpha, mmax);
        k_alpha_exp<<<(NEF + 255) / 256, 256, 0, stream>>>(ei, alpha, mmax, denom);

        hipMemsetAsync(h, 0, (size_t)NNODE * HD * sizeof(float), stream);
        k_message<<<NEF, 256, 0, stream>>>(ei, alpha, denom, xw, h);
        k_ln_elu<<<NNODE, 256, 0, stream>>>(h, bb, lng, lnb);
    }

    // --- readout: h_node = h @ Wr + br (f32 WMMA) ---
    {
        const float* Wr = (const float*)d_in[40];
        const float* br = (const float*)d_in[41];
        dim3 grid(HIDD / 128, (NNODE + 63) / 64);
        k_gemm_f32<HD, HIDD><<<grid, 256, 0, stream>>>(h, Wr, br, h_node, NNODE, 0);
    }

    // --- pooling ---
    hipMemsetAsync(gsum, 0, (size_t)NGR * HIDD * sizeof(float), stream);
    hipMemsetAsync(gcnt, 0, (size_t)NGR * sizeof(float), stream);
    k_fill<<<((size_t)NGR * HIDD + 255) / 256, 256, 0, stream>>>(gmax, -FLT_MAX, (size_t)NGR * HIDD);
    k_pool<<<((size_t)NNODE * HIDD + 255) / 256, 256, 0, stream>>>(h_node, batch, gsum, gmax, gcnt);
    k_pool_fin<<<((size_t)NGR * HIDD + 255) / 256, 256, 0, stream>>>(gsum, gmax, gcnt, gcat);

    // --- MLP head (f32 WMMA) + L2 normalize ---
    {
        const float* Wp1 = (const float*)d_in[42];
        const float* bp1 = (const float*)d_in[43];
        dim3 grid(HIDD / 128, (NGR + 63) / 64);
        k_gemm_f32<2 * HIDD, HIDD><<<grid, 256, 0, stream>>>(gcat, Wp1, bp1, g1, NGR, 1);
    }
    {
        const float* Wp2 = (const float*)d_in[44];
        const float* bp2 = (const float*)d_in[45];
        dim3 grid(OUTD / 128, (NGR + 63) / 64);
        k_gemm_f32<HIDD, OUTD><<<grid, 256, 0, stream>>>(g1, Wp2, bp2, g_out, NGR, 0);
    }
    k_l2norm<<<NGR, 256, 0, stream>>>(g_out);
}